// ConvSE3_74406013435996
// MI455X (gfx1250) — compile-verified
//
#include <hip/hip_runtime.h>
#include <hip/hip_bf16.h>

typedef __attribute__((ext_vector_type(2))) float v2f;
typedef __attribute__((ext_vector_type(8))) float v8f;

#define N_NODES 8192
#define N_EDGES 32768
#define TE 16          // edges per workgroup tile

// ---------------------------------------------------------------------------
// Kernel 1: radial MLP  h2[e][32] = relu(relu(ef@W1+b1)@W2+b2)
// ---------------------------------------------------------------------------
__global__ __launch_bounds__(256) void radial_mlp_kernel(
    const float* __restrict__ ef,   // [E,2]
    const float* __restrict__ W1,   // [2,32]
    const float* __restrict__ b1,   // [32]
    const float* __restrict__ W2,   // [32,32]
    const float* __restrict__ b2,   // [32]
    float* __restrict__ h2out)      // [E,32]
{
    __shared__ float sW1[64];
    __shared__ float sB1[32];
    __shared__ float sW2[1024];
    __shared__ float sB2[32];
    const int tid = threadIdx.x;
    for (int i = tid; i < 64; i += 256) sW1[i] = W1[i];
    if (tid < 32) { sB1[tid] = b1[tid]; sB2[tid] = b2[tid]; }
    for (int i = tid; i < 1024; i += 256) sW2[i] = W2[i];
    __syncthreads();

    const long long e = (long long)blockIdx.x * 256 + tid;
    const float x0 = ef[e * 2 + 0];
    const float x1 = ef[e * 2 + 1];

    float h1[32];
#pragma unroll
    for (int m = 0; m < 32; ++m) {
        float s = fmaf(x0, sW1[m], fmaf(x1, sW1[32 + m], sB1[m]));
        h1[m] = s > 0.0f ? s : 0.0f;
    }
    float* o = h2out + e * 32;
#pragma unroll 4
    for (int m = 0; m < 32; ++m) {
        float s = sB2[m];
#pragma unroll
        for (int k = 0; k < 32; ++k) s = fmaf(h1[k], sW2[k * 32 + m], s);
        o[m] = s > 0.0f ? s : 0.0f;
    }
}

// ---------------------------------------------------------------------------
// Kernel 2: fused basis contraction + per-edge weight GEMM + output matmul
// + scatter-add.  One workgroup (8 waves, 256 threads) handles 16 edges.
// ---------------------------------------------------------------------------
__global__ __launch_bounds__(256) void conv_se3_main_kernel(
    const float* __restrict__ node0,   // [N,16,1]
    const float* __restrict__ node1,   // [N,16,3]
    const float* __restrict__ node2,   // [N,16,5]
    const float* __restrict__ basis,   // [E,9,19,9]
    const float* __restrict__ W3,      // [32,4864]
    const float* __restrict__ h2,      // [E,32]  (from kernel 1, in ws)
    const long long* __restrict__ src, // [E]
    const long long* __restrict__ dst, // [E]
    float* __restrict__ out)           // deg0 [N,16,1] | deg1 [N,16,3] | deg2 [N,16,5]
{
    __shared__ float feats[TE * 16 * 9];      // [e][ci][din]                  9.2 KB
    __shared__ float tmpb[TE * 16 * 176];     // [e][ci][frd] (171 pad 176)  180.2 KB
    __shared__ float rwb[2 * TE * 304];       // [cc][e][f], double co buffer 38.9 KB

    const int tid  = threadIdx.x;
    const int lane = tid & 31;
    const int wave = tid >> 5;
    const int hi   = lane >> 4;   // 0/1 : K-half select for f32 WMMA operands
    const int lo   = lane & 15;   // row (A) / col (B) index
    const long long e0 = (long long)blockIdx.x * TE;

    // ---- gather source-node features: feats[e][ci][0..8] ----
    {
        const int e  = tid >> 4;
        const int ci = tid & 15;
        const long long s = src[e0 + e];
        float* fp = &feats[(e * 16 + ci) * 9];
        fp[0] = node0[s * 16 + ci];
        const float* n1 = &node1[(s * 16 + ci) * 3];
        fp[1] = n1[0]; fp[2] = n1[1]; fp[3] = n1[2];
        const float* n2 = &node2[(s * 16 + ci) * 5];
        fp[4] = n2[0]; fp[5] = n2[1]; fp[6] = n2[2]; fp[7] = n2[3]; fp[8] = n2[4];
    }

    // ---- h2 tile as WMMA A-operand (M = edge row, K = 0..31), reused all block ----
    v2f aH[8];
    {
        const float* hrow = &h2[(e0 + lo) * 32];
#pragma unroll
        for (int kc = 0; kc < 8; ++kc) {
            const int k0 = kc * 4 + 2 * hi;
            aH[kc] = (v2f){hrow[k0], hrow[k0 + 1]};
        }
    }
    const float fKmask = (hi == 0) ? 1.0f : 0.0f;  // zero K=10,11 lanes via mult
    __syncthreads();

    // =========================================================
    // Phase 1: tmp[e] = feats[e](16x9) @ basis[e](9x171), WMMA
    // K padded 9->12 (3 chunks of 4); N padded 171->176.
    // =========================================================
    for (int e = wave * 2; e < wave * 2 + 2; ++e) {
        const float* frow = &feats[(e * 16 + lo) * 9];
        v2f aF0 = (v2f){frow[2 * hi + 0], frow[2 * hi + 1]};
        v2f aF1 = (v2f){frow[4 + 2 * hi + 0], frow[4 + 2 * hi + 1]};
        v2f aF2 = (v2f){frow[8] * fKmask, 0.0f};   // K=8 valid only in low half

        const float* be = &basis[(e0 + e) * 1539];  // 9*19*9
        for (int nc = 0; nc < 11; ++nc) {
            int c = nc * 16 + lo;
            if (c > 170) c = 170;   // clamp into row; padding cols never read
            // all loads unconditional (addresses valid for every lane);
            // K>8 rows zeroed with cndmask-style select, no exec branching
            const float t8 = be[8 * 171 + c];
            v2f b0 = (v2f){be[(2 * hi + 0) * 171 + c], be[(2 * hi + 1) * 171 + c]};
            v2f b1 = (v2f){be[(4 + 2 * hi) * 171 + c], be[(5 + 2 * hi) * 171 + c]};
            v2f b2 = (v2f){t8 * fKmask, 0.0f};

            v8f acc = {};
            acc = __builtin_amdgcn_wmma_f32_16x16x4_f32(false, aF0, false, b0, (short)0, acc, false, false);
            acc = __builtin_amdgcn_wmma_f32_16x16x4_f32(false, aF1, false, b1, (short)0, acc, false, false);
            acc = __builtin_amdgcn_wmma_f32_16x16x4_f32(false, aF2, false, b2, (short)0, acc, false, false);

            const int frd = nc * 16 + lo;
#pragma unroll
            for (int r = 0; r < 8; ++r) {
                const int ci = r + 8 * hi;
                tmpb[(e * 16 + ci) * 176 + frd] = acc[r];
            }
        }
    }
    __syncthreads();

    // =========================================================
    // Loop over co in pairs; per pair:
    //   Phase 2: rw[cc][e][f] = h2[16e x 32] @ W3[32 x 304] (WMMA over edges)
    //   Phase 3: out[e][co][d] = sum_f rw[cc][e][f] * tmp[e][f][d] -> atomic scatter
    // =========================================================
    for (int cp = 0; cp < 8; ++cp) {
        // ---- phase 2: 2 co x 19 n-chunks = 38 WMMA tasks over 8 waves ----
        for (int t = wave; t < 38; t += 8) {
            const int cc = t / 19;            // 0/1 within pair
            const int nc = t % 19;
            const int j  = (cp * 2 + cc) * 304 + nc * 16 + lo;  // W3 column
            v8f acc = {};
#pragma unroll
            for (int kc = 0; kc < 8; ++kc) {
                const int k0 = kc * 4 + 2 * hi;
                v2f b = (v2f){W3[k0 * 4864 + j], W3[(k0 + 1) * 4864 + j]};
                acc = __builtin_amdgcn_wmma_f32_16x16x4_f32(false, aH[kc], false, b, (short)0, acc, false, false);
            }
#pragma unroll
            for (int r = 0; r < 8; ++r) {
                const int e = r + 8 * hi;
                rwb[(cc * TE + e) * 304 + nc * 16 + lo] = acc[r];
            }
        }
        __syncthreads();

        // ---- phase 3: 288 reduction units (2 co x 16 e x 9 d) over 256 threads ----
        for (int u = tid; u < 2 * TE * 9; u += 256) {
            const int cc = u / (TE * 9);
            const int p  = u % (TE * 9);
            const int e  = p / 9;
            const int d  = p % 9;
            const float* rwe = &rwb[(cc * TE + e) * 304];
            const float* te  = &tmpb[e * 16 * 176 + d];
            // 4 rotating accumulators: break the serial FMA dependence chain
            float s0 = 0.0f, s1 = 0.0f, s2 = 0.0f, s3 = 0.0f;
#pragma unroll
            for (int ci = 0; ci < 16; ci += 4) {
                const float* rc = &rwe[ci * 19];
                const float* tc = &te[ci * 176];
#pragma unroll
                for (int fr = 0; fr < 19; ++fr) s0 = fmaf(rc[fr +  0], tc[fr * 9 +   0], s0);
#pragma unroll
                for (int fr = 0; fr < 19; ++fr) s1 = fmaf(rc[fr + 19], tc[fr * 9 + 176], s1);
#pragma unroll
                for (int fr = 0; fr < 19; ++fr) s2 = fmaf(rc[fr + 38], tc[fr * 9 + 352], s2);
#pragma unroll
                for (int fr = 0; fr < 19; ++fr) s3 = fmaf(rc[fr + 57], tc[fr * 9 + 528], s3);
            }
            const float s = (s0 + s1) + (s2 + s3);
            const int co = cp * 2 + cc;
            const long long n = dst[e0 + e];
            float* ptr;
            if (d == 0)      ptr = out + (n * 16 + co);                                        // deg0
            else if (d < 4)  ptr = out + (long long)N_NODES * 16 + (n * 16 + co) * 3 + (d - 1); // deg1
            else             ptr = out + (long long)N_NODES * 64 + (n * 16 + co) * 5 + (d - 4); // deg2
            atomicAdd(ptr, s);
        }
        __syncthreads();
    }
}

// ---------------------------------------------------------------------------
extern "C" void kernel_launch(void* const* d_in, const int* in_sizes, int n_in,
                              void* d_out, int out_size, void* d_ws, size_t ws_size,
                              hipStream_t stream) {
    (void)in_sizes; (void)n_in; (void)ws_size;
    const float* node0 = (const float*)d_in[0];
    const float* node1 = (const float*)d_in[1];
    const float* node2 = (const float*)d_in[2];
    const float* ef0   = (const float*)d_in[3];
    const float* basis = (const float*)d_in[4];
    const float* W1    = (const float*)d_in[5];
    const float* b1    = (const float*)d_in[6];
    const float* W2    = (const float*)d_in[7];
    const float* b2    = (const float*)d_in[8];
    const float* W3    = (const float*)d_in[9];
    const long long* src = (const long long*)d_in[10];
    const long long* dst = (const long long*)d_in[11];
    float* out = (float*)d_out;
    float* h2  = (float*)d_ws;   // [E,32] scratch = 4 MB

    hipMemsetAsync(d_out, 0, (size_t)out_size * sizeof(float), stream);

    radial_mlp_kernel<<<N_EDGES / 256, 256, 0, stream>>>(ef0, W1, b1, W2, b2, h2);

    conv_se3_main_kernel<<<N_EDGES / TE, 256, 0, stream>>>(
        node0, node1, node2, basis, W3, h2, src, dst, out);
}